// Malis_2267742732361
// MI455X (gfx1250) — compile-verified
//
#include <hip/hip_runtime.h>
#include <stdint.h>

// ---- problem constants (shape fixed by reference: aff (3,8,96,96), gt (8,96,96)) ----
#define Zd 8
#define Yd 96
#define Xd 96
#define NVOX (Zd*Yd*Xd)          // 73728 voxels
#define NEDGE (3*NVOX)           // 221184 edges
#define SORT_N 262144            // next pow2 >= NEDGE (2^18)
#define NLAB 64

typedef __attribute__((ext_vector_type(2))) float v2f;
typedef __attribute__((ext_vector_type(8))) float v8f;

// ---------------------------------------------------------------------------
// edge decode: edge e in [0,3N): d = e/N, voxel i = e%N; +1 neighbor along d
// (wrapped like np.roll; 'valid' false on the last slice along d)
// ---------------------------------------------------------------------------
__device__ __forceinline__ void decode_edge(int e, int& u, int& v, bool& valid) {
  int d = e / NVOX;
  int i = e - d * NVOX;
  int z = i / (Yd * Xd);
  int rem = i - z * (Yd * Xd);
  int y = rem / Xd;
  int x = rem - y * Xd;
  u = i;
  if (d == 0) { valid = (z != Zd - 1); int zn = (z + 1 == Zd) ? 0 : z + 1; v = zn * (Yd * Xd) + y * Xd + x; }
  else if (d == 1) { valid = (y != Yd - 1); int yn = (y + 1 == Yd) ? 0 : y + 1; v = z * (Yd * Xd) + yn * Xd + x; }
  else { valid = (x != Xd - 1); int xn = (x + 1 == Xd) ? 0 : x + 1; v = z * (Yd * Xd) + y * Xd + xn; }
}

// float -> u32 monotone (ascending), then invert for descending sort; tie-break
// by ascending edge index in the low 18 bits => matches stable argsort(-aff).
__device__ __forceinline__ unsigned long long packKey(float f, int e) {
  unsigned b = __float_as_uint(f);
  unsigned m = (b & 0x80000000u) ? ~b : (b | 0x80000000u); // ascending order map
  unsigned k = ~m;                                        // descending
  return ((unsigned long long)k << 32) | (unsigned)e;
}

__device__ __forceinline__ long long shflxor_ll(long long v, int m) {
  int lo = (int)((unsigned long long)v & 0xFFFFFFFFull);
  int hi = (int)(v >> 32);
  lo = __shfl_xor(lo, m, 32);
  hi = __shfl_xor(hi, m, 32);
  return ((long long)hi << 32) | (unsigned long long)(unsigned)lo;
}

// ---------------------------------------------------------------------------
// 1) build constrained-affinity sort keys for both passes + zero edge weights
// ---------------------------------------------------------------------------
__global__ void build_keys(const float* __restrict__ aff, const int* __restrict__ gt,
                           unsigned long long* __restrict__ keyPos,
                           unsigned long long* __restrict__ keyNeg,
                           float* __restrict__ wPos, float* __restrict__ wNeg) {
  int t = blockIdx.x * blockDim.x + threadIdx.x;
  if (t >= SORT_N) return;
  if (t >= NEDGE) { keyPos[t] = ~0ULL; keyNeg[t] = ~0ULL; return; }
  wPos[t] = 0.f;
  wNeg[t] = 0.f;
  int u, v; bool valid;
  decode_edge(t, u, v, valid);
  float a = aff[t];
  int gu = gt[u], gv = gt[v];
  float gaff = (gu == gv && gu > 0) ? 1.f : 0.f;
  float ninf = -__builtin_inff();
  float ap = valid ? fminf(a, gaff) : ninf;   // constrained positive pass
  float an = valid ? fmaxf(a, gaff) : ninf;   // constrained negative pass
  keyPos[t] = packKey(ap, t);
  keyNeg[t] = packKey(an, t);
}

// ---------------------------------------------------------------------------
// 2) per-voxel one-hot label histograms (label 0 = background zeroed)
// ---------------------------------------------------------------------------
__global__ void init_counts(const int* __restrict__ gt, int* __restrict__ cPos,
                            int* __restrict__ cNeg) {
  int t = blockIdx.x * blockDim.x + threadIdx.x;
  if (t >= NVOX * NLAB) return;
  int l = t & (NLAB - 1);
  int i = t >> 6;
  int val = (gt[i] == l && l > 0) ? 1 : 0;
  cPos[t] = val;
  cNeg[t] = val;
}

// ---------------------------------------------------------------------------
// 3) bitonic sort pass on both key arrays at once (ascending u64)
// ---------------------------------------------------------------------------
__global__ void bitonic_pass(unsigned long long* a0, unsigned long long* a1,
                             unsigned k, unsigned j) {
  unsigned t = blockIdx.x * blockDim.x + threadIdx.x;
  unsigned long long* d = (t < SORT_N) ? a0 : a1;
  unsigned i = t & (SORT_N - 1);
  unsigned ixj = i ^ j;
  if (ixj > i) {
    unsigned long long vi = d[i], vj = d[ixj];
    bool up = ((i & k) == 0);
    if ((vi > vj) == up) { d[i] = vj; d[ixj] = vi; }
  }
}

// ---------------------------------------------------------------------------
// 4) serial Kruskal/union-find MALIS scan: one wave32 per pass.
//    parent[73728] lives entirely in LDS (288KB of the 320KB/WGP on CDNA5).
//    Each lane owns 2 of the 64 labels: histogram dot + merge is one b64
//    load per root per lane; scalar pair counts via shfl-xor reductions.
// ---------------------------------------------------------------------------
__global__ void __launch_bounds__(32) malis_scan(
    const unsigned long long* __restrict__ keyPos,
    const unsigned long long* __restrict__ keyNeg,
    int* cPos, int* cNeg, float* wPos, float* wNeg) {
  extern __shared__ int parent[];
  const bool isNeg = (blockIdx.x != 0);
  const unsigned long long* __restrict__ key = isNeg ? keyNeg : keyPos;
  int* cnt = isNeg ? cNeg : cPos;
  float* w = isNeg ? wNeg : wPos;
  const int lane = threadIdx.x;

  for (int i = lane; i < NVOX; i += 32) parent[i] = i;
  __syncthreads();

  for (int t = 0; t < NEDGE; ++t) {
    if ((t & 63) == 0)  // stream the sorted edge list ahead of the scan
      __builtin_prefetch((const void*)(key + t + 256), 0, 0);
    unsigned long long p = key[t];
    int e = (int)(p & (unsigned long long)(SORT_N - 1));
    int u, v; bool valid;
    decode_edge(e, u, v, valid);
    if (!valid) { u = 0; v = 0; }  // boundary edges -> self loop, never merge

    int ra = u; for (;;) { int q = parent[ra]; if (q == ra) break; ra = q; }
    int rb = v; for (;;) { int q = parent[rb]; if (q == rb) break; rb = q; }
    if (ra == rb) continue;  // uniform across the wave

    int2 ca = *(const int2*)(cnt + (size_t)ra * NLAB + 2 * lane);
    int2 cb = *(const int2*)(cnt + (size_t)rb * NLAB + 2 * lane);
    long long pp = (long long)ca.x * cb.x + (long long)ca.y * cb.y;
    // pack (sa,sb) into one 64b value; totals <= 73728 so no carry between halves
    long long sab = ((long long)(ca.x + ca.y) << 32) | (unsigned)(cb.x + cb.y);
    for (int m = 16; m > 0; m >>= 1) {
      pp  += shflxor_ll(pp, m);
      sab += shflxor_ll(sab, m);
    }
    int sa = (int)(sab >> 32);
    int sb = (int)(sab & 0xFFFFFFFFll);
    long long pairs = isNeg ? ((long long)sa * sb - pp) : pp;

    int big   = (sa >= sb) ? ra : rb;   // union by foreground size
    int small = (sa >= sb) ? rb : ra;
    if (lane == 0) {
      parent[small] = big;
      w[e] = (float)pairs;
    }
    int2 s2; s2.x = ca.x + cb.x; s2.y = ca.y + cb.y;
    *(int2*)(cnt + (size_t)big * NLAB + 2 * lane) = s2;
    int2 z2; z2.x = 0; z2.y = 0;
    *(int2*)(cnt + (size_t)small * NLAB + 2 * lane) = z2;
  }
}

// ---------------------------------------------------------------------------
// 5) grad = wPos*(1-a) - wNeg*a, summed over all edges.
//    Summation via V_WMMA_F32_16X16X4_F32: A = 64 gradients (16x4, 2 VGPRs per
//    lane), B = ones => each D row replicates the row-sum; accumulate in C,
//    fold the C tile at the end (fp32 FMA accumulation throughout).
// ---------------------------------------------------------------------------
__global__ void __launch_bounds__(256) grad_reduce(
    const float* __restrict__ aff, const float* __restrict__ wPos,
    const float* __restrict__ wNeg, float* __restrict__ out) {
  __shared__ float partial[8];
  const int lane = threadIdx.x & 31;
  const int wv = threadIdx.x >> 5;   // 8 waves per block
  v8f c = {0.f, 0.f, 0.f, 0.f, 0.f, 0.f, 0.f, 0.f};
  v2f ones = {1.f, 1.f};
  const int nchunks = NEDGE / 64;    // 3456, divides exactly
  for (int ch = wv; ch < nchunks; ch += 8) {
    int base = ch * 64 + 2 * lane;
    float a0 = aff[base], a1 = aff[base + 1];
    float g0 = wPos[base] * (1.f - a0) - wNeg[base] * a0;
    float g1 = wPos[base + 1] * (1.f - a1) - wNeg[base + 1] * a1;
    v2f av = {g0, g1};
#if __has_builtin(__builtin_amdgcn_wmma_f32_16x16x4_f32)
    c = __builtin_amdgcn_wmma_f32_16x16x4_f32(
        /*neg_a=*/false, av, /*neg_b=*/false, ones,
        /*c_mod=*/(short)0, c, /*reuse_a=*/false, /*reuse_b=*/false);
#else
    c[0] += g0 + g1;  // scalar fallback (keeps compile alive if builtin absent)
#endif
  }
  // lanes 0..15 hold rows M=0..7 of C, lanes 16..31 rows M=8..15 (col-replicated)
  float s = c[0] + c[1] + c[2] + c[3] + c[4] + c[5] + c[6] + c[7];
  s += __shfl_xor(s, 16, 32);
  if (lane == 0) partial[wv] = s;
  __syncthreads();
  if (threadIdx.x == 0) {
    float tot = 0.f;
    for (int i = 0; i < 8; ++i) tot += partial[i];
    *out = tot;
  }
}

// ---------------------------------------------------------------------------
extern "C" void kernel_launch(void* const* d_in, const int* in_sizes, int n_in,
                              void* d_out, int out_size, void* d_ws, size_t ws_size,
                              hipStream_t stream) {
  (void)in_sizes; (void)n_in; (void)out_size; (void)ws_size;
  const float* aff = (const float*)d_in[0];
  const int* gt = (const int*)d_in[1];

  char* ws = (char*)d_ws;
  size_t off = 0;
  auto carve = [&](size_t bytes) -> void* {
    void* p = (void*)(ws + off);
    off += (bytes + 255) & ~(size_t)255;
    return p;
  };
  unsigned long long* keyPos = (unsigned long long*)carve((size_t)SORT_N * 8);
  unsigned long long* keyNeg = (unsigned long long*)carve((size_t)SORT_N * 8);
  int* cPos  = (int*)carve((size_t)NVOX * NLAB * 4);
  int* cNeg  = (int*)carve((size_t)NVOX * NLAB * 4);
  float* wPos = (float*)carve((size_t)NEDGE * 4);
  float* wNeg = (float*)carve((size_t)NEDGE * 4);

  build_keys<<<SORT_N / 256, 256, 0, stream>>>(aff, gt, keyPos, keyNeg, wPos, wNeg);
  init_counts<<<(NVOX * NLAB) / 256, 256, 0, stream>>>(gt, cPos, cNeg);

  // full bitonic sort of both key arrays (2^18 padded): 171 passes, L2-resident
  for (unsigned k = 2; k <= SORT_N; k <<= 1)
    for (unsigned j = k >> 1; j > 0; j >>= 1)
      bitonic_pass<<<(2 * SORT_N) / 256, 256, 0, stream>>>(keyPos, keyNeg, k, j);

  // two independent serial passes, one WGP each; parent array in LDS (288KB)
  malis_scan<<<2, 32, NVOX * 4, stream>>>(keyPos, keyNeg, cPos, cNeg, wPos, wNeg);

  grad_reduce<<<1, 256, 0, stream>>>(aff, wPos, wNeg, (float*)d_out);
}